// FeedForwardQuantum_65481071396435
// MI455X (gfx1250) — compile-verified
//
#include <hip/hip_runtime.h>
#include <hip/hip_bf16.h>

#define NQ 8
#define QDEPTH 2

// ---------------------------------------------------------------------------
// Kernel 1: 8-qubit statevector simulation, one wave32 per token.
// State: 256 complex amps. index i = (j<<5) | lane ; j = per-lane register 0..7.
// Wire w corresponds to index bit b = 7-w (wire 0 is the most significant axis).
// Bits 5..7 are "register bits" (intra-lane), bits 0..4 are "lane bits" (shfl).
// ---------------------------------------------------------------------------

__device__ __forceinline__ void apply_rx(float (&re)[8], float (&im)[8],
                                         int b, float c, float s, int lane) {
  // RX(theta): symmetric 2x2 -> new = c*mine - i*s*other (same formula both sides)
  if (b >= 5) {
    int m = 1 << (b - 5);
    float nr[8], ni[8];
#pragma unroll
    for (int j = 0; j < 8; ++j) {
      int jp = j ^ m;
      nr[j] = c * re[j] + s * im[jp];
      ni[j] = c * im[j] - s * re[jp];
    }
#pragma unroll
    for (int j = 0; j < 8; ++j) { re[j] = nr[j]; im[j] = ni[j]; }
  } else {
    int lm = 1 << b;
#pragma unroll
    for (int j = 0; j < 8; ++j) {
      float orr = __shfl_xor(re[j], lm, 32);
      float oii = __shfl_xor(im[j], lm, 32);
      re[j] = c * re[j] + s * oii;
      im[j] = c * im[j] - s * orr;
    }
  }
}

__device__ __forceinline__ void apply_rz(float (&re)[8], float (&im)[8],
                                         int b, float c, float s, int lane) {
  // bit==0: *(c - i s); bit==1: *(c + i s)
#pragma unroll
  for (int j = 0; j < 8; ++j) {
    int bit = (b >= 5) ? ((j >> (b - 5)) & 1) : ((lane >> b) & 1);
    float sg = bit ? -s : s;
    float r = re[j], i = im[j];
    re[j] = c * r + sg * i;
    im[j] = c * i - sg * r;
  }
}

__device__ __forceinline__ void apply_cnot(float (&re)[8], float (&im)[8],
                                           int cb, int tb, int lane) {
  // new[i] = (bit_cb(i)==1) ? old[i ^ (1<<tb)] : old[i]
  if (tb >= 5) {
    int tm = 1 << (tb - 5);
    if (cb >= 5) {
#pragma unroll
      for (int j = 0; j < 8; ++j) {
        if (((j >> (cb - 5)) & 1) && !((j >> (tb - 5)) & 1)) {
          int jp = j ^ tm;
          float t0 = re[j]; re[j] = re[jp]; re[jp] = t0;
          float t1 = im[j]; im[j] = im[jp]; im[jp] = t1;
        }
      }
    } else {
      int ctl = (lane >> cb) & 1;
#pragma unroll
      for (int j = 0; j < 8; ++j) {
        if (!((j >> (tb - 5)) & 1)) {
          int jp = j ^ tm;
          float a0 = re[j], a1 = re[jp];
          re[j]  = ctl ? a1 : a0;
          re[jp] = ctl ? a0 : a1;
          a0 = im[j]; a1 = im[jp];
          im[j]  = ctl ? a1 : a0;
          im[jp] = ctl ? a0 : a1;
        }
      }
    }
  } else {
    int lm = 1 << tb;
#pragma unroll
    for (int j = 0; j < 8; ++j) {
      int ctl = (cb >= 5) ? ((j >> (cb - 5)) & 1) : ((lane >> cb) & 1);
      float orr = __shfl_xor(re[j], lm, 32);
      float oii = __shfl_xor(im[j], lm, 32);
      re[j] = ctl ? orr : re[j];
      im[j] = ctl ? oii : im[j];
    }
  }
}

__global__ __launch_bounds__(256) void quantum_layer_kernel(
    const float* __restrict__ x,     // [ntok][embed]
    const float* __restrict__ prx,   // [DEPTH][NQ]
    const float* __restrict__ prz,   // [DEPTH][NQ]
    float* __restrict__ qout,        // [ntok][NQ]
    int ntok, int embed) {
  const int lane = threadIdx.x & 31;
  const int wid = (int)((blockIdx.x * (unsigned)blockDim.x + threadIdx.x) >> 5);
  if (wid >= ntok) return;

  float re[8], im[8];
#pragma unroll
  for (int j = 0; j < 8; ++j) { re[j] = 0.f; im[j] = 0.f; }
  re[0] = (lane == 0) ? 1.f : 0.f;  // |0...0>

  const float* xr = x + (size_t)wid * embed;
#pragma unroll
  for (int w = 0; w < NQ; ++w) {
    float th = 0.5f * xr[w];
    apply_rx(re, im, 7 - w, cosf(th), sinf(th), lane);
  }
#pragma unroll
  for (int l = 0; l < QDEPTH; ++l) {
#pragma unroll
    for (int w = 0; w < NQ; ++w) {
      float tx = 0.5f * prx[l * NQ + w];
      apply_rx(re, im, 7 - w, cosf(tx), sinf(tx), lane);
      float tz = 0.5f * prz[l * NQ + w];
      apply_rz(re, im, 7 - w, cosf(tz), sinf(tz), lane);
    }
#pragma unroll
    for (int w = 0; w < NQ - 1; ++w)
      apply_cnot(re, im, 7 - w, 7 - (w + 1), lane);   // CNOT(w, w+1)
    apply_cnot(re, im, 0, 7, lane);                   // CNOT(7, 0)
  }

  // <Z_w> = sum_i |amp_i|^2 * (1 - 2*bit_{7-w}(i))
  float p[8], tot = 0.f;
#pragma unroll
  for (int j = 0; j < 8; ++j) { p[j] = re[j] * re[j] + im[j] * im[j]; tot += p[j]; }

  float zv[8];
#pragma unroll
  for (int w = 0; w < NQ; ++w) {
    int b = 7 - w;
    float acc;
    if (b >= 5) {
      acc = 0.f;
#pragma unroll
      for (int j = 0; j < 8; ++j)
        acc += ((j >> (b - 5)) & 1) ? -p[j] : p[j];
    } else {
      acc = ((lane >> b) & 1) ? -tot : tot;
    }
#pragma unroll
    for (int off = 16; off >= 1; off >>= 1)
      acc += __shfl_xor(acc, off, 32);
    zv[w] = acc;
  }
  if (lane == 0) {
#pragma unroll
    for (int w = 0; w < NQ; ++w) qout[(size_t)wid * NQ + w] = zv[w];
  }
}

// ---------------------------------------------------------------------------
// Kernel 2: fused FFN  out = relu(qout@W1^T + b1) @ W2^T + b2
// GEMM2 via V_WMMA_F32_16X16X4_F32. Tile: M=128 x N=64 per workgroup,
// 8 waves; wave w owns rows [w*16, w*16+16) x all 64 cols (4 accum tiles).
// Both A and B tiles are stored K-contiguous in LDS so every WMMA fragment
// is a single aligned ds_load_b64 into an even VGPR pair.
// ---------------------------------------------------------------------------

typedef __attribute__((ext_vector_type(2))) float v2f;
typedef __attribute__((ext_vector_type(8))) float v8f;

#define TM 128
#define TN 64
#define TK 32
#define LDK 34   // padded K stride (floats) for both LDS tiles

__global__ __launch_bounds__(256) void ffn_wmma_kernel(
    const float* __restrict__ qout,  // [M][NQ]
    const float* __restrict__ W1,    // [F][NQ]
    const float* __restrict__ b1,    // [F]
    const float* __restrict__ W2,    // [E][F]
    const float* __restrict__ b2,    // [E]
    float* __restrict__ out,         // [M][E]
    int M, int F, int E) {
  __shared__ float qs[TM * NQ];       // 4 KB   q tile
  __shared__ float sA[TM * LDK];      // 17 KB  relu(h) tile, [m][k], K-contig
  __shared__ float sB[TN * LDK];      // 8.7 KB W2 tile,      [n][k], K-contig

  const int t = threadIdx.x;
  const int lane = t & 31;
  const int wid = t >> 5;             // wave 0..7
  const int m0 = blockIdx.y * TM;
  const int n0 = blockIdx.x * TN;

  // preload q tile: 128 rows x 8
#pragma unroll
  for (int r = 0; r < (TM * NQ) / 256; ++r) {
    int L = r * 256 + t;
    qs[L] = qout[(size_t)m0 * NQ + L];
  }
  __syncthreads();

  v8f acc[4];
#pragma unroll
  for (int nt = 0; nt < 4; ++nt)
#pragma unroll
    for (int v = 0; v < 8; ++v) acc[nt][v] = 0.f;

  const int half = lane >> 4;         // 0|1 -> selects K pair within x4 step
  const int l16 = lane & 15;
  const int mrow = wid * 16 + l16;    // A-fragment row in tile

  for (int kc = 0; kc < F; kc += TK) {
    // ---- build relu(h) tile: sA[i][j] = relu(b1[f] + sum_q qs[i][q]*W1[f][q])
#pragma unroll
    for (int r = 0; r < (TM * TK) / 256; ++r) {
      int L = r * 256 + t;            // consecutive t -> consecutive j
      int i = L >> 5;
      int j = L & 31;
      int f = kc + j;
      const float* qrow = &qs[i * NQ];
      const float* wrow = &W1[(size_t)f * NQ];
      float h = b1[f];
#pragma unroll
      for (int q = 0; q < NQ; ++q) h = fmaf(qrow[q], wrow[q], h);
      sA[i * LDK + j] = (h > 0.f) ? h : 0.f;
    }
    // ---- stage W2 tile transposed-in-LDS: sB[n][j] = W2[(n0+n)*F + kc + j]
    //      (global read coalesced along j; LDS write conflict-free)
#pragma unroll
    for (int r = 0; r < (TK * TN) / 256; ++r) {
      int L = r * 256 + t;
      int j = L & 31;
      int n = L >> 5;
      const float* src = &W2[(size_t)(n0 + n) * F + (kc + j)];
      sB[n * LDK + j] = *src;
      // pull next K chunk of this row toward the WGP while WMMAs run
      if (kc + TK < F) __builtin_prefetch(src + TK, 0, 0);
    }
    __syncthreads();

    // ---- 8 x WMMA F32 16x16x4 per accumulator tile over this K chunk
#pragma unroll
    for (int k4 = 0; k4 < TK; k4 += 4) {
      const int ka = k4 + half * 2;   // even -> 8B-aligned LDS loads
      v2f a = *(const v2f*)&sA[mrow * LDK + ka];
#pragma unroll
      for (int nt = 0; nt < 4; ++nt) {
        const int n = nt * 16 + l16;
        v2f b = *(const v2f*)&sB[n * LDK + ka];
        acc[nt] = __builtin_amdgcn_wmma_f32_16x16x4_f32(
            /*neg_a=*/false, a, /*neg_b=*/false, b,
            /*c_mod=*/(short)0, acc[nt], /*reuse_a=*/false, /*reuse_b=*/false);
      }
    }
    __syncthreads();
  }

  // ---- epilogue: + b2, store. C layout: VGPR v -> M = v + 8*half, N = l16.
#pragma unroll
  for (int nt = 0; nt < 4; ++nt) {
    int n = n0 + nt * 16 + l16;
    float bias = b2[n];
#pragma unroll
    for (int v = 0; v < 8; ++v) {
      int m = m0 + wid * 16 + half * 8 + v;
      out[(size_t)m * E + n] = acc[nt][v] + bias;
    }
  }
}

// ---------------------------------------------------------------------------

extern "C" void kernel_launch(void* const* d_in, const int* in_sizes, int n_in,
                              void* d_out, int out_size, void* d_ws, size_t ws_size,
                              hipStream_t stream) {
  const float* x   = (const float*)d_in[0];  // [4,2048,1024]
  const float* prx = (const float*)d_in[1];  // [2,8]
  const float* prz = (const float*)d_in[2];  // [2,8]
  const float* W1  = (const float*)d_in[3];  // [4096,8]
  const float* b1  = (const float*)d_in[4];  // [4096]
  const float* W2  = (const float*)d_in[5];  // [1024,4096]
  const float* b2  = (const float*)d_in[6];  // [1024]
  float* out = (float*)d_out;

  const int E = 1024;
  const int F = 4096;
  const int M = in_sizes[0] / E;             // 8192 tokens

  float* qout = (float*)d_ws;                // [M][8] scratch (256 KB)

  // Kernel 1: one wave32 per token
  int blocks1 = (M * 32 + 255) / 256;
  quantum_layer_kernel<<<blocks1, 256, 0, stream>>>(x, prx, prz, qout, M, E);

  // Kernel 2: fused FFN with WMMA
  dim3 grid2(E / TN, M / TM);
  ffn_wmma_kernel<<<grid2, 256, 0, stream>>>(qout, W1, b1, W2, b2, out, M, F, E);
}